// VarlenPooler_16020228014424
// MI455X (gfx1250) — compile-verified
//
#include <hip/hip_runtime.h>

// VarlenPooler: out[i, :] = x[offsets[i+1] - 1, :]  (last row of each segment)
// Pure gather: 0 FLOPs, ~32 MB of HBM traffic -> bandwidth-bound (~1.4 us at
// 23.3 TB/s). Strategy: one float4 (B128) per thread, 64 threads (2 wave32s)
// per 1 KB row, nontemporal hints on the streaming load/store (gfx1250 TH
// cache controls), huge grid for memory-level parallelism.

typedef __attribute__((ext_vector_type(4))) float v4f;

__global__ __launch_bounds__(256) void varlen_last_gather(
    const float* __restrict__ x,          // [total_tokens, 256] f32
    const int* __restrict__ offsets,      // [num_segments + 1] i32
    float* __restrict__ out,              // [num_segments, 256] f32
    int num_segments)
{
    const int tid = blockIdx.x * 256 + threadIdx.x;
    const int row = tid >> 6;             // 64 float4's per 256-float row
    if (row >= num_segments) return;
    const int c4  = tid & 63;             // float4 column within the row

    // Source row index for this segment (all 64 threads of a row read the
    // same offset; it lives in L2/WGP$ after the first touch).
    const long long src_row = (long long)offsets[row + 1] - 1;

    // 64-bit pointer math: x spans 1 GB (2^28 floats), byte offsets exceed 2^31.
    const v4f* __restrict__ src = (const v4f*)(x + src_row * 256LL) + c4;
    v4f v = __builtin_nontemporal_load(src);          // global_load_b128 th:NT

    v4f* __restrict__ dst = (v4f*)(out + (long long)row * 256LL) + c4;
    __builtin_nontemporal_store(v, dst);              // global_store_b128 th:NT
}

extern "C" void kernel_launch(void* const* d_in, const int* in_sizes, int n_in,
                              void* d_out, int out_size, void* d_ws, size_t ws_size,
                              hipStream_t stream) {
    const float* x       = (const float*)d_in[0];   // [TOTAL_TOKENS * 256] f32
    const int*   offsets = (const int*)d_in[1];     // [NUM_SEGMENTS + 1] i32
    float*       out     = (float*)d_out;           // [NUM_SEGMENTS * 256] f32

    const int num_segments = in_sizes[1] - 1;       // 16384
    // 64 float4 slots per row, one per thread.
    const long long threads_total = (long long)num_segments * 64;
    const int blocks = (int)((threads_total + 255) / 256);

    varlen_last_gather<<<blocks, 256, 0, stream>>>(x, offsets, out, num_segments);
}